// Group_26104811225234
// MI455X (gfx1250) — compile-verified
//
#include <hip/hip_runtime.h>
#include <hip/hip_bf16.h>

typedef float v2f __attribute__((ext_vector_type(2)));
typedef float v8f __attribute__((ext_vector_type(8)));

#define FPS1_N      32768
#define FPS1_NP     4096
#define FPS2_N      4096
#define FPS2_NP     1024
#define GROUPS      1024
#define GSIZE       64
#define TILE_Q      16          // queries per KNN block (one WMMA M-tile)

// ---------------------------------------------------------------------------
// argmax/argmin combine helpers (tie -> lower index, matches jnp.argmax/top_k)
// ---------------------------------------------------------------------------
__device__ __forceinline__ void combine_max(float& bd, int& bi, float od, int oi) {
    if (od > bd || (od == bd && oi < bi)) { bd = od; bi = oi; }
}
__device__ __forceinline__ void combine_min(float& bd, int& bi, float od, int oi) {
    if (od < bd || (od == bd && oi < bi)) { bd = od; bi = oi; }
}

// ---------------------------------------------------------------------------
// FPS stage 1: N=32768 -> 4096 samples. One 1024-thread block per batch.
// dists live in VGPRs (32/thread); points stream from L2 each iteration
// (384KB/batch resident in 192MB L2). Three barriers per iteration:
// wave32 shuffle-butterfly -> 32-entry LDS -> wave0 butterfly.
// ---------------------------------------------------------------------------
__global__ __launch_bounds__(1024)
void fps_stage1(const float* __restrict__ xyz, float* __restrict__ centers1) {
    const int b    = blockIdx.x;
    const int tid  = threadIdx.x;
    const int lane = tid & 31;
    const int wave = tid >> 5;

    const float* pts = xyz + (size_t)b * FPS1_N * 3;
    float*       ctr = centers1 + (size_t)b * FPS1_NP * 3;

    float dist[32];
#pragma unroll
    for (int j = 0; j < 32; ++j) dist[j] = 1e10f;

    __shared__ float s_c[3];
    __shared__ float s_bd[32];
    __shared__ int   s_bi[32];
    __shared__ int   s_far;
    if (tid == 0) s_far = 0;
    __syncthreads();

    for (int i = 0; i < FPS1_NP; ++i) {
        if (tid == 0) {
            const int f = s_far;
            const float cx = pts[f * 3 + 0], cy = pts[f * 3 + 1], cz = pts[f * 3 + 2];
            ctr[i * 3 + 0] = cx; ctr[i * 3 + 1] = cy; ctr[i * 3 + 2] = cz;
            s_c[0] = cx; s_c[1] = cy; s_c[2] = cz;
        }
        __syncthreads();
        const float cx = s_c[0], cy = s_c[1], cz = s_c[2];

        float bd = -1.0f; int bi = 0x7fffffff;
#pragma unroll
        for (int j = 0; j < 32; ++j) {
            const int p = j * 1024 + tid;
            const float dx = pts[p * 3 + 0] - cx;
            const float dy = pts[p * 3 + 1] - cy;
            const float dz = pts[p * 3 + 2] - cz;
            const float d  = dx * dx + dy * dy + dz * dz;
            const float nd = fminf(dist[j], d);
            dist[j] = nd;
            combine_max(bd, bi, nd, p);
        }
        // wave32 butterfly
#pragma unroll
        for (int off = 16; off > 0; off >>= 1)
            combine_max(bd, bi, __shfl_xor(bd, off, 32), __shfl_xor(bi, off, 32));
        if (lane == 0) { s_bd[wave] = bd; s_bi[wave] = bi; }
        __syncthreads();
        if (wave == 0) {
            bd = s_bd[lane]; bi = s_bi[lane];
#pragma unroll
            for (int off = 16; off > 0; off >>= 1)
                combine_max(bd, bi, __shfl_xor(bd, off, 32), __shfl_xor(bi, off, 32));
            if (lane == 0) s_far = bi;
        }
        __syncthreads();
    }
}

// ---------------------------------------------------------------------------
// FPS stage 2: N=4096 -> 1024 samples, point set cached in LDS (48KB << 320KB).
// Writes centers2 to workspace AND the `center` output region of d_out.
// ---------------------------------------------------------------------------
__global__ __launch_bounds__(1024)
void fps_stage2(const float* __restrict__ centers1, float* __restrict__ centers2,
                float* __restrict__ center_out) {
    const int b    = blockIdx.x;
    const int tid  = threadIdx.x;
    const int lane = tid & 31;
    const int wave = tid >> 5;

    const float* src = centers1 + (size_t)b * FPS2_N * 3;
    float*       c2  = centers2 + (size_t)b * FPS2_NP * 3;
    float*       co  = center_out + (size_t)b * FPS2_NP * 3;

    __shared__ float s_pts[FPS2_N * 3];   // 48 KB
    __shared__ float s_c[3];
    __shared__ float s_bd[32];
    __shared__ int   s_bi[32];
    __shared__ int   s_far;

    for (int t = tid; t < FPS2_N * 3; t += 1024) s_pts[t] = src[t];
    if (tid == 0) s_far = 0;
    __syncthreads();

    float dist[4];
#pragma unroll
    for (int j = 0; j < 4; ++j) dist[j] = 1e10f;

    for (int i = 0; i < FPS2_NP; ++i) {
        if (tid == 0) {
            const int f = s_far;
            const float cx = s_pts[f * 3 + 0], cy = s_pts[f * 3 + 1], cz = s_pts[f * 3 + 2];
            c2[i * 3 + 0] = cx; c2[i * 3 + 1] = cy; c2[i * 3 + 2] = cz;
            co[i * 3 + 0] = cx; co[i * 3 + 1] = cy; co[i * 3 + 2] = cz;
            s_c[0] = cx; s_c[1] = cy; s_c[2] = cz;
        }
        __syncthreads();
        const float cx = s_c[0], cy = s_c[1], cz = s_c[2];

        float bd = -1.0f; int bi = 0x7fffffff;
#pragma unroll
        for (int j = 0; j < 4; ++j) {
            const int p = j * 1024 + tid;
            const float dx = s_pts[p * 3 + 0] - cx;
            const float dy = s_pts[p * 3 + 1] - cy;
            const float dz = s_pts[p * 3 + 2] - cz;
            const float d  = dx * dx + dy * dy + dz * dz;
            const float nd = fminf(dist[j], d);
            dist[j] = nd;
            combine_max(bd, bi, nd, p);
        }
#pragma unroll
        for (int off = 16; off > 0; off >>= 1)
            combine_max(bd, bi, __shfl_xor(bd, off, 32), __shfl_xor(bi, off, 32));
        if (lane == 0) { s_bd[wave] = bd; s_bi[wave] = bi; }
        __syncthreads();
        if (wave == 0) {
            bd = s_bd[lane]; bi = s_bi[lane];
#pragma unroll
            for (int off = 16; off > 0; off >>= 1)
                combine_max(bd, bi, __shfl_xor(bd, off, 32), __shfl_xor(bi, off, 32));
            if (lane == 0) s_far = bi;
        }
        __syncthreads();
    }
}

// ---------------------------------------------------------------------------
// KNN + gather. One 256-thread block per (batch, 16-query tile).
// Distances via V_WMMA_F32_16X16X4_F32:
//   A row m  = (-2 q.x, -2 q.y, -2 q.z, 1)      (16x4)
//   B col n  = ( p.x,    p.y,    p.z,  |p|^2 )  (4x16)
//   D[m][n]  = |p_n|^2 - 2 q_m . p_n            (|q|^2 row-constant dropped:
//                                                does not change top-k order)
// Operand construction is BRANCH-FREE (all lanes load all 3 components and
// compute |p|^2, then v_cndmask-select per lane half) so EXEC stays all-1s
// through the WMMA loop (ISA requirement) and no saveexec/branch pairs are
// emitted per chunk.
// 16 x 4096 distance tile lives in 256KB dynamic LDS (CDNA5: 320KB/WGP).
// Selection: wave w owns queries 2w/2w+1; 64 rounds of wave32 argmin with
// +INF marking (same-wave DS ops are in-order -> no barrier needed).
// ---------------------------------------------------------------------------
__global__ __launch_bounds__(256)
void knn_group(const float* __restrict__ pts1,   // centers1 (B,4096,3)
               const float* __restrict__ ctr2,   // centers2 (B,1024,3)
               float* __restrict__ nbr) {        // (B,1024,64,3)
    extern __shared__ float s_dist[];            // TILE_Q * 4096 floats

    const int b     = blockIdx.x >> 6;           // 64 tiles per batch
    const int tile  = blockIdx.x & 63;
    const int gbase = tile * TILE_Q;
    const int tid   = threadIdx.x;
    const int lane  = tid & 31;
    const int wave  = tid >> 5;
    const bool lo   = (lane < 16);               // lane half select (K=0,1 vs K=2,3)

    const float* P = pts1 + (size_t)b * FPS2_N * 3;
    const float* Q = ctr2 + ((size_t)b * GROUPS + gbase) * 3;

    // A operand: 16x4 f32 -> 2 VGPRs. lanes 0-15: K=0,1 ; lanes 16-31: K=2,3
    const int m = lane & 15;
    const float qx = Q[m * 3 + 0], qy = Q[m * 3 + 1], qz = Q[m * 3 + 2];
    v2f a;
    a.x = lo ? (-2.0f * qx) : (-2.0f * qz);
    a.y = lo ? (-2.0f * qy) : 1.0f;

    // ---- distance tile via WMMA: each wave does 32 of the 256 N-chunks ----
    for (int t = 0; t < 32; ++t) {
        const int c = wave + 8 * t;              // chunk of 16 points
        const int n = c * 16 + (lane & 15);
        // unconditional 3-component load + norm: no exec divergence
        const float px  = P[n * 3 + 0];
        const float py  = P[n * 3 + 1];
        const float pz  = P[n * 3 + 2];
        const float nrm = px * px + py * py + pz * pz;
        v2f bb;
        bb.x = lo ? px : pz;
        bb.y = lo ? py : nrm;

        v8f acc = {};
        acc = __builtin_amdgcn_wmma_f32_16x16x4_f32(
                  false, a, false, bb, (short)0, acc, false, false);

#pragma unroll
        for (int v = 0; v < 8; ++v) {
            const int row = v + 8 * (lane >> 4);
            const int col = c * 16 + (lane & 15);
            s_dist[row * FPS2_N + col] = acc[v];
        }
    }
    __syncthreads();

    // ---- per-query top-64 selection + gather/center ----
    for (int qq = 0; qq < 2; ++qq) {
        const int   q    = wave * 2 + qq;
        float*      rowD = s_dist + q * FPS2_N;
        const float cqx = Q[q * 3 + 0], cqy = Q[q * 3 + 1], cqz = Q[q * 3 + 2];
        float* out = nbr + ((size_t)b * GROUPS + gbase + q) * GSIZE * 3;

        for (int k = 0; k < GSIZE; ++k) {
            float bd = 3.4e38f; int bi = 0x7fffffff;
#pragma unroll
            for (int j = 0; j < FPS2_N / 32; ++j) {
                const int n = j * 32 + lane;     // conflict-free lane stride
                combine_min(bd, bi, rowD[n], n);
            }
#pragma unroll
            for (int off = 16; off > 0; off >>= 1)
                combine_min(bd, bi, __shfl_xor(bd, off, 32), __shfl_xor(bi, off, 32));
            // butterfly converges: all lanes hold the same (bd, bi)
            if (lane == 0) {
                out[k * 3 + 0] = P[bi * 3 + 0] - cqx;
                out[k * 3 + 1] = P[bi * 3 + 1] - cqy;
                out[k * 3 + 2] = P[bi * 3 + 2] - cqz;
                rowD[bi] = 3.4e38f;              // exclude from later rounds
            }
        }
    }
}

// ---------------------------------------------------------------------------
extern "C" void kernel_launch(void* const* d_in, const int* in_sizes, int n_in,
                              void* d_out, int out_size, void* d_ws, size_t ws_size,
                              hipStream_t stream) {
    const float* xyz = (const float*)d_in[0];
    const int B = in_sizes[0] / (FPS1_N * 3);    // = 8

    float* nbr_out = (float*)d_out;                                   // B*1024*64*3
    float* ctr_out = (float*)d_out + (size_t)B * GROUPS * GSIZE * 3;  // B*1024*3

    float* centers1 = (float*)d_ws;                                   // B*4096*3
    float* centers2 = centers1 + (size_t)B * FPS2_N * 3;              // B*1024*3

    fps_stage1<<<B, 1024, 0, stream>>>(xyz, centers1);
    fps_stage2<<<B, 1024, 0, stream>>>(centers1, centers2, ctr_out);

    const size_t knn_lds = (size_t)TILE_Q * FPS2_N * sizeof(float);   // 256 KB
    knn_group<<<B * (GROUPS / TILE_Q), 256, knn_lds, stream>>>(centers1, centers2, nbr_out);
}